// NetGIN_27178553049848
// MI455X (gfx1250) — compile-verified
//
#include <hip/hip_runtime.h>
#include <stdint.h>

#define N_NODES 32768
#define N_EDGES 524288
#define N_B     64
#define BN_EPS  1e-5f

typedef __attribute__((ext_vector_type(2))) float        v2f;
typedef __attribute__((ext_vector_type(8))) float        v8f;
typedef __attribute__((ext_vector_type(4))) unsigned int u32x4;
typedef __attribute__((ext_vector_type(8))) int          i32x8;
typedef __attribute__((ext_vector_type(4))) int          i32x4;

#define MODE_BNRELU  0
#define MODE_ADD2    1
#define MODE_CONCAT4 2
#define MODE_MAXMEAN 3

__device__ __forceinline__ float4 ld4(const float* p) {
  return *reinterpret_cast<const float4*>(p);
}

// ---------------------------------------------------------------------------
// TDM: stage a rows x cols fp32 tile (row stride = src_stride elements) from
// global into LDS (packed, stride = cols).  D# built per CDNA5 ISA ch.8:
//   group0: count=1 | lds_addr | global_addr[56:0] | type=2
//   group1: data_size=4B, tensor_dim0/1, tile_dim0/1, tensor_dim0_stride
// This toolchain exposes the 6-arg builtin (extra int32x8 group before cpol).
// Issued by wave 0 only; caller waits s_wait_tensorcnt then barriers.
// ---------------------------------------------------------------------------
__device__ __forceinline__ void tdm_load_f32_2d(void* lds_dst, const float* gsrc,
                                                unsigned rows, unsigned cols,
                                                unsigned src_stride) {
  unsigned long long ga = (unsigned long long)(uintptr_t)gsrc;
  unsigned lo = (unsigned)(uintptr_t)lds_dst;   // low 32 bits = LDS byte offset
  u32x4 g0;
  g0[0] = 1u;                                            // count=1, user D#
  g0[1] = lo;                                            // lds_addr
  g0[2] = (unsigned)(ga & 0xffffffffull);                // global_addr lo
  g0[3] = ((unsigned)((ga >> 32) & 0x01ffffffull)) | (2u << 30); // addr hi | type=2
  i32x8 g1;
  g1[0] = (int)(2u << 16);                               // mask=0, data_size=2 (4B)
  g1[1] = (int)((cols & 0xffffu) << 16);                 // tensor_dim0[15:0] @ 63:48
  g1[2] = (int)(((cols >> 16) & 0xffffu) | ((rows & 0xffffu) << 16)); // dim0 hi | dim1 lo
  g1[3] = (int)(((rows >> 16) & 0xffffu) | ((cols & 0xffffu) << 16)); // dim1 hi | tile_dim0
  g1[4] = (int)(rows & 0xffffu);                         // tile_dim1 (tile_dim2=0)
  g1[5] = (int)src_stride;                               // tensor_dim0_stride lo32
  g1[6] = 0;
  g1[7] = 0;
  i32x4 gz4 = {0, 0, 0, 0};
  i32x8 gz8 = {0, 0, 0, 0, 0, 0, 0, 0};
  __builtin_amdgcn_tensor_load_to_lds(g0, g1, gz4, gz4, gz8, 0);
}

// ---------------------------------------------------------------------------
// Wave-level fp32 WMMA 16x16 tile:  acc += A[16 x klen] * B[klen x 16]
// A fragment (16x4 f32): lane<16 -> a={A[m][k],A[m][k+1]}, lane>=16 -> K+2.
// B fragment (4x16 f32): mirrored (K on vgpr+lane-half, N on lane&15).
// ---------------------------------------------------------------------------
__device__ __forceinline__ v8f wmma_kt(const float* As, int lda,
                                       const float* Bs, int ldb,
                                       v8f acc, int klen) {
  const int lane = threadIdx.x & 31;
  const int m  = lane & 15;
  const int kh = (lane >> 4) << 1;
#pragma unroll
  for (int k = 0; k < klen; k += 4) {
    v2f a, b;
    a[0] = As[m * lda + k + kh + 0];
    a[1] = As[m * lda + k + kh + 1];
    b[0] = Bs[(k + kh + 0) * ldb + m];
    b[1] = Bs[(k + kh + 1) * ldb + m];
    acc = __builtin_amdgcn_wmma_f32_16x16x4_f32(false, a, false, b,
                                                (short)0, acc, false, false);
  }
  return acc;
}

// Store 16x16 f32 C/D fragment (+bias), accumulate per-column sum/sumsq in LDS.
__device__ __forceinline__ void tile_epilogue(const v8f& acc, float* out, int ldc,
                                              int growBase, int gcolBase,
                                              const float* bias,
                                              float* sS, float* sQ, int localColBase) {
  const int lane = threadIdx.x & 31;
  const int col = lane & 15;
  const int rh  = (lane >> 4) * 8;
  const int gcol = gcolBase + col;
  const float b = bias[gcol];
  float s = 0.f, q = 0.f;
#pragma unroll
  for (int r = 0; r < 8; ++r) {
    float v = acc[r] + b;
    out[(size_t)(growBase + rh + r) * ldc + gcol] = v;
    s += v; q += v * v;
  }
  atomicAdd(&sS[localColBase + col], s);
  atomicAdd(&sQ[localColBase + col], q);
}

// ---------------------------------------------------------------------------
__global__ __launch_bounds__(256) void k_zero(float* p, long long n) {
  long long i = (long long)blockIdx.x * 256 + threadIdx.x;
  if (i < n) p[i] = 0.f;
}

// scale = g*rsqrt(var+eps), shift = be - mean*scale  (biased var, training BN)
__global__ void k_bn_finalize(const float* stSum, const float* stSq,
                              const float* g, const float* be,
                              float* scale, float* shift, int C, float invn) {
  int c = blockIdx.x * 64 + threadIdx.x;
  if (c < C) {
    float m = stSum[c] * invn;
    float var = stSq[c] * invn - m * m;
    float s = g[c] * rsqrtf(var + BN_EPS);
    scale[c] = s;
    shift[c] = be[c] - m * s;
  }
}

// ---------------------------------------------------------------------------
// EdgeConv layer 1: build m=[x_d, x_s-x_d] (K=8) per edge, y = m@W1+b1 -> ybuf.
// 128 edges/block, 8 waves = 8 M-tiles x 4 N-tiles.  W1 staged by TDM.
// ---------------------------------------------------------------------------
__global__ __launch_bounds__(256) void k_edge_l1(const float* x, const int* ei,
                                                 const float* W1, const float* b1,
                                                 float* ybuf,
                                                 float* stSum, float* stSq) {
  __shared__ float As[128][12];           // K=8 padded to 12 (bank spread)
  __shared__ float Ws[8 * 64];
  __shared__ float sS[64], sQ[64];
  const int t = threadIdx.x;
  if (t < 64) { sS[t] = 0.f; sQ[t] = 0.f; }
  if (t < 32) tdm_load_f32_2d(Ws, W1, 8, 64, 64);
  const int ebase = blockIdx.x * 128;
  if (t < 128) {
    int e = ebase + t;
    int s = ei[e];
    int d = ei[N_EDGES + e];
    float4 xs = ld4(x + (size_t)s * 4);
    float4 xd = ld4(x + (size_t)d * 4);
    As[t][0] = xd.x; As[t][1] = xd.y; As[t][2] = xd.z; As[t][3] = xd.w;
    As[t][4] = xs.x - xd.x; As[t][5] = xs.y - xd.y;
    As[t][6] = xs.z - xd.z; As[t][7] = xs.w - xd.w;
  }
  if (t < 32) __builtin_amdgcn_s_wait_tensorcnt(0);
  __syncthreads();
  const int w = t >> 5;
  const float* Ap = &As[w * 16][0];
  for (int nt = 0; nt < 4; ++nt) {
    v8f acc = {0, 0, 0, 0, 0, 0, 0, 0};
    acc = wmma_kt(Ap, 12, &Ws[nt * 16], 64, acc, 8);
    tile_epilogue(acc, ybuf, 64, ebase + w * 16, nt * 16, b1, sS, sQ, nt * 16);
  }
  __syncthreads();
  if (t < 64) { atomicAdd(&stSum[t], sS[t]); atomicAdd(&stSq[t], sQ[t]); }
}

// ---------------------------------------------------------------------------
// Unified K-tiled GEMM: out[M,Cout] = A[M,K] @ W[K,Cout] + bias, stats per col.
// 64 rows x 64 cols per block; A-chunk (64x64) rebuilt per K-tile by mode-
// specific loader (fused BN+ReLU / GIN add / concat / max|mean concat).
// W tile staged into LDS with the Tensor Data Mover each K-tile.
// ---------------------------------------------------------------------------
__global__ __launch_bounds__(256) void k_gemm(
    int mode,
    const float* p0, const float* p1, const float* p2, const float* p3,
    const float* scale, const float* shift,
    const float* W, const float* bias,
    float* out, int K, int Cout,
    float* stSum, float* stSq) {
  __shared__ float As[64][68];
  __shared__ float Ws[64 * 64];
  __shared__ float sS[64], sQ[64];
  const int t = threadIdx.x;
  if (t < 64) { sS[t] = 0.f; sQ[t] = 0.f; }
  const int rowBase = blockIdx.x * 64;
  const int colBase = blockIdx.y * 64;
  const int w   = t >> 5;
  const int mt  = w >> 1;
  const int ntb = (w & 1) * 2;
  v8f acc0 = {0, 0, 0, 0, 0, 0, 0, 0};
  v8f acc1 = {0, 0, 0, 0, 0, 0, 0, 0};
  const int nkt = K >> 6;
  for (int kt = 0; kt < nkt; ++kt) {
    const int f0 = kt * 64;
    __syncthreads();                       // previous tile fully consumed
    if (t < 32)
      tdm_load_f32_2d(Ws, W + (size_t)f0 * Cout + colBase, 64, 64, (unsigned)Cout);
    {
      const int row = t >> 2;
      const int seg = t & 3;
      const size_t grow = (size_t)rowBase + row;
      for (int j = 0; j < 4; ++j) {
        const int c = (seg * 4 + j) * 4;
        const int f = f0 + c;
        float4 v;
        if (mode == MODE_ADD2) {
          float4 a = ld4(p0 + grow * K + f);
          float4 g = ld4(p1 + grow * K + f);
          v = make_float4(a.x + g.x, a.y + g.y, a.z + g.z, a.w + g.w);
        } else if (mode == MODE_CONCAT4) {
          if (f < 64)       v = ld4(p0 + grow * 64 + f);
          else if (f < 128) v = ld4(p1 + grow * 64 + (f - 64));
          else if (f < 256) v = ld4(p2 + grow * 128 + (f - 128));
          else              v = ld4(p3 + grow * 256 + (f - 256));
        } else if (mode == MODE_MAXMEAN) {
          v = (f < 1024) ? ld4(p0 + grow * 1024 + f)
                         : ld4(p1 + grow * 1024 + (f - 1024));
        } else { // MODE_BNRELU
          v = ld4(p0 + grow * K + f);
          v.x = fmaxf(fmaf(v.x, scale[f + 0], shift[f + 0]), 0.f);
          v.y = fmaxf(fmaf(v.y, scale[f + 1], shift[f + 1]), 0.f);
          v.z = fmaxf(fmaf(v.z, scale[f + 2], shift[f + 2]), 0.f);
          v.w = fmaxf(fmaf(v.w, scale[f + 3], shift[f + 3]), 0.f);
        }
        As[row][c + 0] = v.x; As[row][c + 1] = v.y;
        As[row][c + 2] = v.z; As[row][c + 3] = v.w;
      }
    }
    if (t < 32) __builtin_amdgcn_s_wait_tensorcnt(0);
    __syncthreads();
    acc0 = wmma_kt(&As[mt * 16][0], 68, &Ws[ntb * 16], 64, acc0, 64);
    acc1 = wmma_kt(&As[mt * 16][0], 68, &Ws[(ntb + 1) * 16], 64, acc1, 64);
  }
  tile_epilogue(acc0, out, Cout, rowBase + mt * 16, colBase + ntb * 16,
                bias, sS, sQ, ntb * 16);
  tile_epilogue(acc1, out, Cout, rowBase + mt * 16, colBase + (ntb + 1) * 16,
                bias, sS, sQ, (ntb + 1) * 16);
  __syncthreads();
  if (t < 64) {
    atomicAdd(&stSum[colBase + t], sS[t]);
    atomicAdd(&stSq[colBase + t], sQ[t]);
  }
}

// EdgeConv reduce: x0[dst] = max over edges of relu(bn(y)).  Post-ReLU values
// are >=0 so int-compare atomicMax on float bits is exact; zero-init buffer
// also reproduces the reference's empty-segment -> 0 behaviour.
__global__ __launch_bounds__(256) void k_edge_max(const float* y, const int* ei,
                                                  const float* scale, const float* shift,
                                                  float* x0) {
  long long idx = (long long)blockIdx.x * 256 + threadIdx.x;
  int c = (int)(idx & 63);
  long long e = idx >> 6;
  float v = fmaxf(fmaf(y[idx], scale[c], shift[c]), 0.f);
  int d = ei[N_EDGES + e];
  atomicMax((int*)x0 + (size_t)d * 64 + c, __float_as_int(v));
}

// GIN neighbor sum: agg[dst] += h[src]
__global__ __launch_bounds__(256) void k_scatter(const float* h, const int* ei,
                                                 float* agg, int logC) {
  long long idx = (long long)blockIdx.x * 256 + threadIdx.x;
  int C = 1 << logC;
  long long e = idx >> logC;
  int c = (int)(idx & (C - 1));
  int s = ei[e];
  int d = ei[N_EDGES + e];
  atomicAdd(&agg[((size_t)d << logC) + c], h[((size_t)s << logC) + c]);
}

// t = relu(in*scale+shift) -> out, accumulate per-channel stats (C = 64/128/256)
__global__ __launch_bounds__(256) void k_bnrelu_stats(const float* in, float* outp,
                                                      int C,
                                                      const float* scale, const float* shift,
                                                      float* stSum, float* stSq) {
  __shared__ float sS[256], sQ[256];
  const int t = threadIdx.x;
  if (t < C) { sS[t] = 0.f; sQ[t] = 0.f; }
  __syncthreads();
  long long idx = (long long)blockIdx.x * 256 + t;
  int c = (int)(idx & (C - 1));
  float v = fmaxf(fmaf(in[idx], scale[c], shift[c]), 0.f);
  outp[idx] = v;
  atomicAdd(&sS[c], v);
  atomicAdd(&sQ[c], v * v);
  __syncthreads();
  if (t < C) { atomicAdd(&stSum[t], sS[t]); atomicAdd(&stSq[t], sQ[t]); }
}

// in-place affine (GIN outer BatchNorm)
__global__ __launch_bounds__(256) void k_affine(float* p, int C,
                                                const float* scale, const float* shift) {
  long long idx = (long long)blockIdx.x * 256 + threadIdx.x;
  int c = (int)(idx & (C - 1));
  p[idx] = fmaf(p[idx], scale[c], shift[c]);
}

// Graph pooling: each thread owns one (graph, channel); 512 contiguous nodes
// per graph, so no atomics are needed.  h = relu(bn(lin_out)) fused here.
__global__ __launch_bounds__(256) void k_pool(const float* hbuf,
                                              const float* scale, const float* shift,
                                              float* gmax, float* gmean) {
  int g = blockIdx.x;
  int c = blockIdx.y * 256 + threadIdx.x;
  const float s = scale[c], sh = shift[c];
  const float* base = hbuf + (size_t)g * 512 * 1024 + c;
  float mx = 0.f, sm = 0.f;   // post-ReLU values >= 0
  for (int r = 0; r < 512; ++r) {
    float v = fmaxf(fmaf(base[(size_t)r * 1024], s, sh), 0.f);
    mx = fmaxf(mx, v);
    sm += v;
  }
  gmax[g * 1024 + c]  = mx;
  gmean[g * 1024 + c] = sm * (1.f / 512.f);
}

// Final 256->40 linear + log_softmax, one block per batch row (tiny).
__global__ void k_head(const float* h2buf, const float* scale, const float* shift,
                       const float* Wo, const float* bo, float* outp) {
  __shared__ float o[64];
  __shared__ float red[2];
  const int row = blockIdx.x;
  const int t = threadIdx.x;
  float acc = 0.f;
  if (t < 40) {
    for (int k = 0; k < 256; ++k) {
      float a = fmaxf(fmaf(h2buf[row * 256 + k], scale[k], shift[k]), 0.f);
      acc = fmaf(a, Wo[k * 40 + t], acc);
    }
    acc += bo[t];
    o[t] = acc;
  }
  __syncthreads();
  if (t == 0) {
    float m = o[0];
    for (int i = 1; i < 40; ++i) m = fmaxf(m, o[i]);
    float se = 0.f;
    for (int i = 0; i < 40; ++i) se += expf(o[i] - m);
    red[0] = m;
    red[1] = logf(se);
  }
  __syncthreads();
  if (t < 40) outp[row * 40 + t] = acc - red[0] - red[1];
}

// ---------------------------------------------------------------------------
extern "C" void kernel_launch(void* const* d_in, const int* in_sizes, int n_in,
                              void* d_out, int out_size, void* d_ws, size_t ws_size,
                              hipStream_t stream) {
  (void)in_sizes; (void)n_in; (void)out_size; (void)ws_size;
  const float* x  = (const float*)d_in[0];
  const int*   ei = (const int*)d_in[1];
#define FIN(i) ((const float*)d_in[i])

  // --- workspace layout (fp32 elements) ---
  float* f = (float*)d_ws;
  float* ybuf  = f; f += (size_t)N_EDGES * 64;   // edge MLP activations (in-place)
  float* x0    = f; f += (size_t)N_NODES * 64;
  float* x1    = f; f += (size_t)N_NODES * 64;
  float* x2    = f; f += (size_t)N_NODES * 128;
  float* x3    = f; f += (size_t)N_NODES * 256;
  float* agg   = f; f += (size_t)N_NODES * 128;  // reused per GIN layer
  float* ybn   = f; f += (size_t)N_NODES * 256;  // GIN inner linear out (reused)
  float* hbuf  = f; f += (size_t)N_NODES * 1024; // lin linear out
  float* gmaxb = f; f += (size_t)N_B * 1024;
  float* gmnb  = f; f += (size_t)N_B * 1024;
  float* h1buf = f; f += (size_t)N_B * 512;
  float* h2buf = f; f += (size_t)N_B * 256;
  float* stSum = f; f += 1024;
  float* stSq  = f; f += 1024;                   // contiguous with stSum
  float* scA   = f; f += 1024;
  float* shA   = f; f += 1024;
  float* scB   = f; f += 1024;
  float* shB   = f; f += 1024;

  const float invE = 1.f / (float)N_EDGES;
  const float invN = 1.f / (float)N_NODES;
  const float invB = 1.f / (float)N_B;

#define ZERO(p, n) k_zero<<<(unsigned)(((n) + 255) / 256), 256, 0, stream>>>((p), (long long)(n))
#define ZSTATS() ZERO(stSum, 2048)

  // ---------------- EdgeConv: MLP(8->64->64->64), max over dst ----------------
  ZSTATS();
  k_edge_l1<<<N_EDGES / 128, 256, 0, stream>>>(x, ei, FIN(3), FIN(4), ybuf, stSum, stSq);
  k_bn_finalize<<<1, 64, 0, stream>>>(stSum, stSq, FIN(5), FIN(6), scA, shA, 64, invE);

  ZSTATS();
  k_gemm<<<dim3(N_EDGES / 64, 1), 256, 0, stream>>>(MODE_BNRELU, ybuf, nullptr, nullptr, nullptr,
      scA, shA, FIN(7), FIN(8), ybuf, 64, 64, stSum, stSq);
  k_bn_finalize<<<1, 64, 0, stream>>>(stSum, stSq, FIN(9), FIN(10), scA, shA, 64, invE);

  ZSTATS();
  k_gemm<<<dim3(N_EDGES / 64, 1), 256, 0, stream>>>(MODE_BNRELU, ybuf, nullptr, nullptr, nullptr,
      scA, shA, FIN(11), FIN(12), ybuf, 64, 64, stSum, stSq);
  k_bn_finalize<<<1, 64, 0, stream>>>(stSum, stSq, FIN(13), FIN(14), scA, shA, 64, invE);

  ZERO(x0, (size_t)N_NODES * 64);
  k_edge_max<<<(unsigned)((size_t)N_EDGES * 64 / 256), 256, 0, stream>>>(ybuf, ei, scA, shA, x0);

  // ---------------- GIN layers ----------------
  struct GinCfg { const float* h; float* xo; int Cin, logC, Cout;
                  int Wi, bi, gi, bei, og, obe; };
  GinCfg gins[3] = {
    { x0, x1,  64, 6,  64, 15, 16, 17, 18, 27, 28 },
    { x1, x2,  64, 6, 128, 19, 20, 21, 22, 29, 30 },
    { x2, x3, 128, 7, 256, 23, 24, 25, 26, 31, 32 },
  };
  for (int L = 0; L < 3; ++L) {
    GinCfg& G = gins[L];
    ZERO(agg, (size_t)N_NODES * G.Cin);
    k_scatter<<<(unsigned)((size_t)N_EDGES * G.Cin / 256), 256, 0, stream>>>(G.h, ei, agg, G.logC);
    ZSTATS();
    k_gemm<<<dim3(N_NODES / 64, G.Cout / 64), 256, 0, stream>>>(MODE_ADD2, G.h, agg, nullptr, nullptr,
        nullptr, nullptr, FIN(G.Wi), FIN(G.bi), ybn, G.Cin, G.Cout, stSum, stSq);
    k_bn_finalize<<<(G.Cout + 63) / 64, 64, 0, stream>>>(stSum, stSq, FIN(G.gi), FIN(G.bei),
        scA, shA, G.Cout, invN);
    ZSTATS();
    k_bnrelu_stats<<<(unsigned)((size_t)N_NODES * G.Cout / 256), 256, 0, stream>>>(
        ybn, G.xo, G.Cout, scA, shA, stSum, stSq);
    k_bn_finalize<<<(G.Cout + 63) / 64, 64, 0, stream>>>(stSum, stSq, FIN(G.og), FIN(G.obe),
        scB, shB, G.Cout, invN);
    k_affine<<<(unsigned)((size_t)N_NODES * G.Cout / 256), 256, 0, stream>>>(G.xo, G.Cout, scB, shB);
  }

  // ---------------- lin: concat[512] -> 1024, then pooling ----------------
  ZSTATS();
  k_gemm<<<dim3(N_NODES / 64, 1024 / 64), 256, 0, stream>>>(MODE_CONCAT4, x0, x1, x2, x3,
      nullptr, nullptr, FIN(33), FIN(34), hbuf, 512, 1024, stSum, stSq);
  k_bn_finalize<<<16, 64, 0, stream>>>(stSum, stSq, FIN(35), FIN(36), scA, shA, 1024, invN);
  k_pool<<<dim3(N_B, 4), 256, 0, stream>>>(hbuf, scA, shA, gmaxb, gmnb);

  // ---------------- head: 2048 -> 512 -> 256 -> 40 + log_softmax ----------------
  ZSTATS();
  k_gemm<<<dim3(1, 512 / 64), 256, 0, stream>>>(MODE_MAXMEAN, gmaxb, gmnb, nullptr, nullptr,
      nullptr, nullptr, FIN(37), FIN(38), h1buf, 2048, 512, stSum, stSq);
  k_bn_finalize<<<8, 64, 0, stream>>>(stSum, stSq, FIN(39), FIN(40), scA, shA, 512, invB);

  ZSTATS();
  k_gemm<<<dim3(1, 256 / 64), 256, 0, stream>>>(MODE_BNRELU, h1buf, nullptr, nullptr, nullptr,
      scA, shA, FIN(41), FIN(42), h2buf, 512, 256, stSum, stSq);
  k_bn_finalize<<<4, 64, 0, stream>>>(stSum, stSq, FIN(43), FIN(44), scB, shB, 256, invB);

  k_head<<<N_B, 64, 0, stream>>>(h2buf, scB, shB, FIN(45), FIN(46), (float*)d_out);
}